// SimpleSpectralConv2D_38680475468341
// MI455X (gfx1250) — compile-verified
//
#include <hip/hip_runtime.h>
#include <hip/hip_bf16.h>

// CDNA5 / gfx1250 implementation of SimpleSpectralConv2D.
//
// Math:  out[f,o,c] = relu( Q[f,o,c] - Lout[o] * P[f,o,c] )
//        Q[f,o,c]   = sum_t SM[f,t] * Lin[h(o,t)] * x[h(o,t), c]
//        P[f,o,c]   = sum_t SM[f,t] *               x[h(o,t), c]
//        h(o,t)     = (oy + t/3)*64 + (ox + t%3),  o = oy*62 + ox, t = 0..8
//
// GEMM mapping (per tile): D(16x16) = A(16x20) @ B(20x16) via five
// V_WMMA_F32_16X16X4_F32 (K=20 in 4-steps).  A rows 0-7 hold SM on K 0-8
// (Q path), rows 8-15 hold SM on K 9-17 (P path), K 18-19 zero pad.
// B rows 0-8 = Lin-scaled x taps, rows 9-17 = raw x taps, N = 16 channels.
// D rows 0-7 = Q(f=row), rows 8-15 = P(f=row-8); paired via shfl_xor(16).

typedef __attribute__((ext_vector_type(2))) float v2f;
typedef __attribute__((ext_vector_type(8))) float v8f;

#define F_    8
#define W_    64
#define C_    64
#define OUT_H 62
#define OUT_W 62
#define L_OUT (OUT_H * OUT_W)     // 3844
#define NTILES (L_OUT * 4)        // 15376 = 1922 blocks * 8 waves

__global__ __launch_bounds__(256)
void spectral_conv_wmma(const float* __restrict__ x,     // [4096, 64]
                        const float* __restrict__ SM,    // [8, 9]
                        const float* __restrict__ Lin,   // [4096]
                        const float* __restrict__ Lout,  // [3844]
                        float* __restrict__ out)         // [8, 62, 62, 64]
{
    const int lane = threadIdx.x & 31;
    const int wave = threadIdx.x >> 5;
    const int tile = blockIdx.x * 8 + wave;        // grid sized exactly: no guard,
                                                   // EXEC stays all-ones for WMMA
    const int o  = tile >> 2;                      // output position
    const int c0 = (tile & 3) << 4;                // channel tile base
    const int oy = o / OUT_W;
    const int ox = o - oy * OUT_W;

    const int n    = lane & 15;                    // N index (channel offset) / A row
    const int hi   = lane >> 4;                    // lane half
    const int koff = hi << 1;                      // K lane-split: {0,1} vs {2,3}
    const int c    = c0 + n;
    const int m    = n;                            // A-matrix row for this lane

    // ---- load the 9 input taps once; reused by both Q (scaled) and P rows ----
    float xv[9], lv[9];
#pragma unroll
    for (int t = 0; t < 9; ++t) {
        const int b = t / 3, j = t - 3 * b;
        const int h = (oy + b) * W_ + (ox + j);
        xv[t] = x[h * C_ + c];                     // 64B coalesced per half-wave
        lv[t] = Lin[h];
    }

    // ---- A fragment: block-diagonal SM_kernel (uniform across whole grid) ----
    auto aval = [&](int k) -> float {
        if (m < 8)  return (k < 9) ? SM[m * 9 + k] : 0.0f;
        const int mm = m - 8;
        return (k >= 9 && k < 18) ? SM[mm * 9 + (k - 9)] : 0.0f;
    };
    // ---- B fragment: im2col rows (0-8 Lin*x, 9-17 x, 18-19 pad) ----
    auto bval = [&](int t) -> float {
        if (t < 9)  return lv[t] * xv[t];
        if (t < 18) return xv[t - 9];
        return 0.0f;
    };

    v8f acc = {0.f, 0.f, 0.f, 0.f, 0.f, 0.f, 0.f, 0.f};
#pragma unroll
    for (int s = 0; s < 5; ++s) {                  // K = 20 in steps of 4
        v2f A, B;
        A.x = aval(4 * s + koff);
        A.y = aval(4 * s + koff + 1);
        B.x = bval(4 * s + koff);
        B.y = bval(4 * s + koff + 1);
        // 8 args: (neg_a, A, neg_b, B, c_mod, C, reuse_a, reuse_b)
        acc = __builtin_amdgcn_wmma_f32_16x16x4_f32(
            false, A, false, B, (short)0, acc, false, false);
    }

    // ---- epilogue: out = relu(Q - Lout[o] * P), pair rows via lane-half swap ----
    const float lsc = Lout[o];
#pragma unroll
    for (int r = 0; r < 8; ++r) {                  // D reg r: M = r + 8*hi, f = r
        const float mine  = acc[r];
        const float other = __shfl_xor(mine, 16, 32);   // wave32 half swap
        const float q = hi ? other : mine;
        const float p = hi ? mine  : other;
        float val = q - lsc * p;
        val = val > 0.0f ? val : 0.0f;
        // lo half stores f = 0..3, hi half stores f = 4..7 (identical values)
        const bool doStore = (r < 4) ? (hi == 0) : (hi == 1);
        if (doStore)
            out[((r * OUT_H + oy) * OUT_W + ox) * C_ + c] = val;
    }
}

extern "C" void kernel_launch(void* const* d_in, const int* in_sizes, int n_in,
                              void* d_out, int out_size, void* d_ws, size_t ws_size,
                              hipStream_t stream) {
    const float* x    = (const float*)d_in[0];   // [1,64,64,64]
    const float* SM   = (const float*)d_in[1];   // [8,9]
    const float* Lin  = (const float*)d_in[2];   // [1,4096]
    const float* Lout = (const float*)d_in[3];   // [3844,1]
    float* out = (float*)d_out;                  // [8,62,62,64]

    const int blocks = NTILES / 8;               // 1922, exact
    spectral_conv_wmma<<<blocks, 256, 0, stream>>>(x, SM, Lin, Lout, out);
}